// DGL_GCN_Test_4810363372757
// MI455X (gfx1250) — compile-verified
//
#include <hip/hip_runtime.h>

typedef float v2f __attribute__((ext_vector_type(2)));
typedef float v8f __attribute__((ext_vector_type(8)));

#define N_NODES 50000
#define N_EDGES 800000
#define N_GRAPH 512
#define F_IN    74
#define HID     64
#define BN_EPS  1e-5f

// ---------------------------------------------------------------------------
// fill kernels (agg / stats / readout init)
// ---------------------------------------------------------------------------
__global__ void fill_f32(float* __restrict__ p, float v, int n) {
    int i = blockIdx.x * blockDim.x + threadIdx.x;
    if (i < n) p[i] = v;
}
__global__ void fill_u32(unsigned* __restrict__ p, unsigned v, int n) {
    int i = blockIdx.x * blockDim.x + threadIdx.x;
    if (i < n) p[i] = v;
}

__device__ __forceinline__ v8f wmma_f32(v2f a, v2f b, v8f c) {
    return __builtin_amdgcn_wmma_f32_16x16x4_f32(false, a, false, b,
                                                 (short)0, c, false, false);
}

// ---------------------------------------------------------------------------
// Dual GEMM: XW = X@W ; RL = relu(X@rW + rb).  One wave = 16 rows x 64 cols.
// V_WMMA_F32_16X16X4_F32, operands staged in LDS in fragment order so the
// hot loop is ds_load_b64 + wmma only (no per-element global loads/waits).
// ---------------------------------------------------------------------------
template<int DIN>
__global__ __launch_bounds__(128)
void gemm2_wmma(const float* __restrict__ X,
                const float* __restrict__ W,    // DIN x 64
                const float* __restrict__ rW,   // DIN x 64
                const float* __restrict__ rb,   // 64
                float* __restrict__ XW,         // nrows x 64
                float* __restrict__ RL,         // nrows x 64
                int ntiles)
{
    constexpr int KT   = (DIN + 3) / 4;   // K chunks of 4
    constexpr int DPAD = KT * 4;          // zero-padded row length

    __shared__ v2f  fragW[KT * 128];      // (kc, tile, lane) B-fragments of W
    __shared__ v2f  fragR[KT * 128];      // same for rW
    __shared__ float xt[4 * 16 * DPAD];   // per-wave 16-row X tile, padded

    const int  lane = threadIdx.x & 31;
    const int  wave = threadIdx.x >> 5;
    const int  l15  = lane & 15;
    const bool hi   = lane >= 16;

    // ---- cooperative B-fragment build (all 128 threads; K-mask baked in) ----
    for (int idx = threadIdx.x; idx < KT * 128; idx += 128) {
        const int kc  = idx >> 7;
        const int r   = idx & 127;
        const int n   = ((r >> 5) << 4) + (r & 15);      // tile*16 + l15
        const int ka0 = kc * 4 + ((r & 31) >= 16 ? 2 : 0);
        const int ka1 = ka0 + 1;
        const int c0  = (ka0 < DIN) ? ka0 : 0;
        const int c1  = (ka1 < DIN) ? ka1 : 0;
        const float m0 = (ka0 < DIN) ? 1.0f : 0.0f;
        const float m1 = (ka1 < DIN) ? 1.0f : 0.0f;
        v2f fw; fw.x = W [(size_t)c0 * HID + n] * m0; fw.y = W [(size_t)c1 * HID + n] * m1;
        v2f fr; fr.x = rW[(size_t)c0 * HID + n] * m0; fr.y = rW[(size_t)c1 * HID + n] * m1;
        fragW[idx] = fw;
        fragR[idx] = fr;
    }
    __syncthreads();

    // ---- per-wave tile (inactive waves clamp + skip store: no barrier/EXEC
    //      hazards, WMMA always runs with EXEC all-1s) ----
    const int  tile  = blockIdx.x * 4 + wave;
    const bool activ = tile < ntiles;
    const int  row0  = (activ ? tile : (ntiles - 1)) * 16;

    float* xw_ = xt + wave * 16 * DPAD;
    for (int r = 0; r < 16; ++r) {
        const float* xrow = X + (size_t)(row0 + r) * DIN;
        for (int k = lane; k < DPAD; k += 32)
            xw_[r * DPAD + k] = (k < DIN) ? xrow[k] : 0.0f;
    }
    // wave-internal LDS ordering: DS ops from one wave are processed in order,
    // so the fragment loads below observe these stores.

    v8f acc[4], racc[4];
    #pragma unroll
    for (int t = 0; t < 4; ++t) { acc[t] = {}; racc[t] = {}; }

    const v2f* arow = (const v2f*)(xw_ + l15 * DPAD);   // DPAD%4==0 -> aligned
    const int  aoff = hi ? 1 : 0;

    #pragma unroll
    for (int kc = 0; kc < KT; ++kc) {
        const v2f a = arow[kc * 2 + aoff];
        #pragma unroll
        for (int t = 0; t < 4; ++t) {
            const v2f b  = fragW[kc * 128 + t * 32 + lane];
            const v2f rv = fragR[kc * 128 + t * 32 + lane];
            acc[t]  = wmma_f32(a, b,  acc[t]);
            racc[t] = wmma_f32(a, rv, racc[t]);
        }
    }

    if (activ) {
        // C/D layout: VGPR j holds row (j + hi*8), col = lane&15 of the tile
        const int rbase = row0 + (hi ? 8 : 0);
        #pragma unroll
        for (int t = 0; t < 4; ++t) {
            const int n = t * 16 + l15;
            const float bias = rb[n];
            #pragma unroll
            for (int j = 0; j < 8; ++j) {
                const int r = rbase + j;
                XW[(size_t)r * HID + n] = acc[t][j];
                float v = racc[t][j] + bias;
                RL[(size_t)r * HID + n] = v > 0.0f ? v : 0.0f;
            }
        }
    }
}

// ---------------------------------------------------------------------------
// Edge scatter: agg[dst] += xw[src].  One thread = one edge x 4 features.
// ---------------------------------------------------------------------------
__global__ __launch_bounds__(256)
void edge_agg(const int* __restrict__ src, const int* __restrict__ dst,
              const float* __restrict__ xw, float* __restrict__ agg, int ne)
{
    int idx = blockIdx.x * blockDim.x + threadIdx.x;
    int e = idx >> 4;
    if (e >= ne) return;
    int g = (idx & 15) << 2;
    int s = src[e], d = dst[e];
    float4 v = *(const float4*)(xw + (size_t)s * HID + g);
    float* ap = agg + (size_t)d * HID + g;
    atomicAdd(ap + 0, v.x);
    atomicAdd(ap + 1, v.y);
    atomicAdd(ap + 2, v.z);
    atomicAdd(ap + 3, v.w);
}

// ---------------------------------------------------------------------------
// h = relu(agg + b) + rl  (in place into agg), and accumulate BN statistics.
// ---------------------------------------------------------------------------
__global__ __launch_bounds__(256)
void combine_stats(float* __restrict__ agg, const float* __restrict__ rl,
                   const float* __restrict__ bvec, float* __restrict__ stats,
                   int nrows)
{
    __shared__ float ssum[256], ssq[256];
    const int f   = threadIdx.x & 63;
    const int sub = threadIdx.x >> 6;
    const int r0  = blockIdx.x * 64;
    const int rend = (r0 + 64 < nrows) ? (r0 + 64) : nrows;
    const float b = bvec[f];
    float s = 0.0f, q = 0.0f;
    for (int r = r0 + sub; r < rend; r += 4) {
        float a = agg[(size_t)r * HID + f] + b;
        a = a > 0.0f ? a : 0.0f;
        float h = a + rl[(size_t)r * HID + f];
        agg[(size_t)r * HID + f] = h;
        s += h; q += h * h;
    }
    ssum[threadIdx.x] = s; ssq[threadIdx.x] = q;
    __syncthreads();
    if (sub == 0) {
        s = ssum[f] + ssum[64 + f] + ssum[128 + f] + ssum[192 + f];
        q = ssq[f]  + ssq[64 + f]  + ssq[128 + f]  + ssq[192 + f];
        atomicAdd(&stats[f], s);
        atomicAdd(&stats[64 + f], q);
    }
}

// ---------------------------------------------------------------------------
// BatchNorm apply (training-mode batch stats, biased variance)
// ---------------------------------------------------------------------------
__global__ __launch_bounds__(256)
void bn_apply(float* __restrict__ h, const float* __restrict__ stats,
              const float* __restrict__ gamma, const float* __restrict__ beta,
              int nrows)
{
    int idx = blockIdx.x * blockDim.x + threadIdx.x;
    if (idx >= nrows * HID) return;
    int f = idx & 63;
    float inv = 1.0f / (float)nrows;
    float mean = stats[f] * inv;
    float var  = stats[64 + f] * inv - mean * mean;
    h[idx] = (h[idx] - mean) * rsqrtf(var + BN_EPS) * gamma[f] + beta[f];
}

// ---------------------------------------------------------------------------
// Readout: gate = sigmoid(h . aw_W + aw_b) via wave32 shuffle reduction;
// segment-sum (atomic f32 add) + segment-max (monotone-uint atomicMax).
// ---------------------------------------------------------------------------
__device__ __forceinline__ unsigned f32_ordered(float f) {
    unsigned b = __float_as_uint(f);
    return (b & 0x80000000u) ? ~b : (b | 0x80000000u);
}
#define NEG_INF_ENC 0x007FFFFFu   // f32_ordered(-inf)

__global__ __launch_bounds__(256)
void readout_nodes(const float* __restrict__ h, const int* __restrict__ gid,
                   const float* __restrict__ awW, const float* __restrict__ awb,
                   float* __restrict__ hsum, unsigned* __restrict__ hmaxu,
                   int nrows)
{
    const int lane = threadIdx.x & 31;
    const int wave = threadIdx.x >> 5;
    const int n = blockIdx.x * 8 + wave;
    if (n >= nrows) return;
    float h0 = h[(size_t)n * HID + lane];
    float h1 = h[(size_t)n * HID + 32 + lane];
    float p  = h0 * awW[lane] + h1 * awW[32 + lane];
    for (int off = 16; off > 0; off >>= 1) p += __shfl_xor(p, off, 32);
    float w = 1.0f / (1.0f + expf(-(p + awb[0])));
    int g = gid[n];
    atomicAdd(&hsum[(size_t)g * HID + lane],      h0 * w);
    atomicAdd(&hsum[(size_t)g * HID + 32 + lane], h1 * w);
    atomicMax(&hmaxu[(size_t)g * HID + lane],      f32_ordered(h0));
    atomicMax(&hmaxu[(size_t)g * HID + 32 + lane], f32_ordered(h1));
}

// Decode max, non-finite -> 0, build hg = [hsum | hmax]  (B x 128)
__global__ __launch_bounds__(256)
void build_hg(const float* __restrict__ hsum, const unsigned* __restrict__ hmaxu,
              float* __restrict__ hg, int nb)
{
    int idx = blockIdx.x * blockDim.x + threadIdx.x;
    if (idx >= nb * HID) return;
    int b = idx >> 6, f = idx & 63;
    hg[(size_t)b * 128 + f] = hsum[idx];
    unsigned u = hmaxu[idx];
    float m = (u & 0x80000000u) ? __uint_as_float(u ^ 0x80000000u)
                                : __uint_as_float(~u);
    if (!(fabsf(m) <= 3.402823466e38f)) m = 0.0f;  // NaN / +-inf -> 0
    hg[(size_t)b * 128 + 64 + f] = m;
}

// ---------------------------------------------------------------------------
// Output projection: out = hg @ out_W + out_b  (512x128 @ 128x64), WMMA f32,
// same LDS fragment staging.  32 tiles -> 8 blocks x 4 waves, all active.
// ---------------------------------------------------------------------------
__global__ __launch_bounds__(128)
void out_gemm_wmma(const float* __restrict__ hg, const float* __restrict__ Wo,
                   const float* __restrict__ bo, float* __restrict__ out)
{
    constexpr int KT = 128 / 4;                 // 32 K-chunks
    __shared__ v2f  fragW[KT * 128];            // 32 KB
    __shared__ float xt[4 * 16 * 128];          // 32 KB

    const int  lane = threadIdx.x & 31;
    const int  wave = threadIdx.x >> 5;
    const int  l15  = lane & 15;
    const bool hi   = lane >= 16;

    for (int idx = threadIdx.x; idx < KT * 128; idx += 128) {
        const int kc  = idx >> 7;
        const int r   = idx & 127;
        const int n   = ((r >> 5) << 4) + (r & 15);
        const int ka0 = kc * 4 + ((r & 31) >= 16 ? 2 : 0);
        v2f fw;
        fw.x = Wo[(size_t)ka0 * HID + n];
        fw.y = Wo[(size_t)(ka0 + 1) * HID + n];
        fragW[idx] = fw;
    }
    __syncthreads();

    const int row0 = (blockIdx.x * 4 + wave) * 16;
    float* xw_ = xt + wave * 16 * 128;
    for (int r = 0; r < 16; ++r) {
        const float* xrow = hg + (size_t)(row0 + r) * 128;
        for (int k = lane; k < 128; k += 32)
            xw_[r * 128 + k] = xrow[k];
    }

    v8f acc[4];
    #pragma unroll
    for (int t = 0; t < 4; ++t) acc[t] = {};

    const v2f* arow = (const v2f*)(xw_ + l15 * 128);
    const int  aoff = hi ? 1 : 0;
    #pragma unroll
    for (int kc = 0; kc < KT; ++kc) {
        const v2f a = arow[kc * 2 + aoff];
        #pragma unroll
        for (int t = 0; t < 4; ++t)
            acc[t] = wmma_f32(a, fragW[kc * 128 + t * 32 + lane], acc[t]);
    }

    const int rbase = row0 + (hi ? 8 : 0);
    #pragma unroll
    for (int t = 0; t < 4; ++t) {
        const int n = t * 16 + l15;
        const float bias = bo[n];
        #pragma unroll
        for (int j = 0; j < 8; ++j)
            out[(size_t)(rbase + j) * HID + n] = acc[t][j] + bias;
    }
}

// ---------------------------------------------------------------------------
// Host driver
// ---------------------------------------------------------------------------
extern "C" void kernel_launch(void* const* d_in, const int* in_sizes, int n_in,
                              void* d_out, int out_size, void* d_ws, size_t ws_size,
                              hipStream_t stream)
{
    (void)in_sizes; (void)n_in; (void)out_size; (void)ws_size;

    const float* feats = (const float*)d_in[0];
    const int*   src   = (const int*)d_in[1];
    const int*   dst   = (const int*)d_in[2];
    const int*   gid   = (const int*)d_in[3];
    const float* awW  = (const float*)d_in[22];
    const float* awb  = (const float*)d_in[23];
    const float* outW = (const float*)d_in[24];
    const float* outb = (const float*)d_in[25];
    float* out = (float*)d_out;

    // workspace layout (floats): xw, rlin, bufA, bufB : N*64 each; stats:128;
    // hsum: B*64; hmaxu: B*64 (u32); hg: B*128.  total ~51.5 MB
    float* ws    = (float*)d_ws;
    float* xw    = ws;
    float* rlin  = xw   + (size_t)N_NODES * HID;
    float* bufA  = rlin + (size_t)N_NODES * HID;
    float* bufB  = bufA + (size_t)N_NODES * HID;
    float* stats = bufB + (size_t)N_NODES * HID;
    float* hsum  = stats + 128;
    unsigned* hmaxu = (unsigned*)(hsum + (size_t)N_GRAPH * HID);
    float* hg    = (float*)(hmaxu + (size_t)N_GRAPH * HID);

    const int nh     = N_NODES * HID;               // 3.2M
    const int ntiles = N_NODES / 16;                // 3125
    const int gemmBlocks = (ntiles + 3) / 4;        // 782

    auto run_layer = [&](const float* x, int din, int pi, float* agg) {
        const float* W  = (const float*)d_in[pi + 0];
        const float* b  = (const float*)d_in[pi + 1];
        const float* rW = (const float*)d_in[pi + 2];
        const float* rb = (const float*)d_in[pi + 3];
        const float* g  = (const float*)d_in[pi + 4];
        const float* be = (const float*)d_in[pi + 5];

        fill_f32<<<(nh + 255) / 256, 256, 0, stream>>>(agg, 0.0f, nh);
        fill_f32<<<1, 128, 0, stream>>>(stats, 0.0f, 128);
        if (din == F_IN)
            gemm2_wmma<F_IN><<<gemmBlocks, 128, 0, stream>>>(x, W, rW, rb, xw, rlin, ntiles);
        else
            gemm2_wmma<HID><<<gemmBlocks, 128, 0, stream>>>(x, W, rW, rb, xw, rlin, ntiles);
        edge_agg<<<(N_EDGES * 16 + 255) / 256, 256, 0, stream>>>(src, dst, xw, agg, N_EDGES);
        combine_stats<<<(N_NODES + 63) / 64, 256, 0, stream>>>(agg, rlin, b, stats, N_NODES);
        bn_apply<<<(nh + 255) / 256, 256, 0, stream>>>(agg, stats, g, be, N_NODES);
    };

    run_layer(feats, F_IN, 4,  bufA);   // h0 in bufA
    run_layer(bufA,  HID, 10, bufB);    // h1 in bufB
    run_layer(bufB,  HID, 16, bufA);    // h2 in bufA

    const int nbh = N_GRAPH * HID;      // 32768
    fill_f32<<<(nbh + 255) / 256, 256, 0, stream>>>(hsum, 0.0f, nbh);
    fill_u32<<<(nbh + 255) / 256, 256, 0, stream>>>(hmaxu, NEG_INF_ENC, nbh);
    readout_nodes<<<(N_NODES + 7) / 8, 256, 0, stream>>>(bufA, gid, awW, awb,
                                                         hsum, hmaxu, N_NODES);
    build_hg<<<(nbh + 255) / 256, 256, 0, stream>>>(hsum, hmaxu, hg, N_GRAPH);
    out_gemm_wmma<<<(N_GRAPH / 16) / 4, 128, 0, stream>>>(hg, outW, outb, out);
}